// GraphConvolution_86517821211632
// MI455X (gfx1250) — compile-verified
//
#include <hip/hip_runtime.h>

typedef __attribute__((ext_vector_type(2))) float v2f;
typedef __attribute__((ext_vector_type(8))) float v8f;

#define NN 100000
#define DD 128

// ---------------------------------------------------------------------------
// Fused dual GEMM: S1 = X @ W1, S2 = X @ W2  (all fp32, exact, via
// V_WMMA_F32_16X16X4_F32). Block = 512 threads = 16 waves. Block stages one
// 16x128 tile of X in LDS; waves 0..7 -> 8 column tiles of W1, waves 8..15 ->
// W2. Each wave owns one 16x16 output tile, K-loop of 32 WMMAs.
// ---------------------------------------------------------------------------
__global__ __launch_bounds__(512) void gemm_dual(const float* __restrict__ X,
                                                 const float* __restrict__ W1,
                                                 const float* __restrict__ W2,
                                                 float* __restrict__ S1,
                                                 float* __restrict__ S2) {
  __shared__ float Atile[16 * DD];  // 8 KB

  const int t = threadIdx.x;
  const int rowBase = blockIdx.x * 16;

  // Cooperative load of the 16x128 A tile: 512 threads x 1 float4 = 2048 floats
  {
    const float4* src = reinterpret_cast<const float4*>(X + (size_t)rowBase * DD);
    float4* dst = reinterpret_cast<float4*>(Atile);
    dst[t] = src[t];
  }
  __syncthreads();

  const int wave = t >> 5;   // 0..15
  const int lane = t & 31;

  const float* __restrict__ W = (wave < 8) ? W1 : W2;
  float* __restrict__ S       = (wave < 8) ? S1 : S2;
  const int colTile = (wave & 7) * 16;

  // ISA 16x4 fp32 A layout: lanes 0-15 hold K=0,1 (v0,v1); lanes 16-31 K=2,3.
  // B (4x16) mirrors: lanes 0-15 rows K=0,1; lanes 16-31 rows K=2,3.
  const int row = lane & 15;           // M for A, N for B/C/D
  const int kb  = (lane >> 4) * 2;     // K sub-offset for this half-wave
  const int col = colTile + row;       // global output column for this lane

  v8f acc = {};
#pragma unroll
  for (int k = 0; k < DD; k += 4) {
    v2f a, b;
    const float2 av =
        *reinterpret_cast<const float2*>(&Atile[row * DD + k + kb]);
    a.x = av.x;
    a.y = av.y;
    b.x = W[(size_t)(k + kb) * DD + col];
    b.y = W[(size_t)(k + kb + 1) * DD + col];
    acc = __builtin_amdgcn_wmma_f32_16x16x4_f32(
        /*neg_a=*/false, a, /*neg_b=*/false, b,
        /*c_mod=*/(short)0, acc, /*reuse_a=*/false, /*reuse_b=*/false);
  }

  // C/D layout: lanes 0-15 -> M = v, lanes 16-31 -> M = v + 8.
  const int mBase = (lane >> 4) * 8;
#pragma unroll
  for (int v = 0; v < 8; ++v) {
    S[(size_t)(rowBase + mBase + v) * DD + col] = acc[v];
  }
}

// ---------------------------------------------------------------------------
// out[n][d] = bias[d]  (also clears the poisoned output buffer)
// ---------------------------------------------------------------------------
__global__ __launch_bounds__(256) void init_out(const float* __restrict__ bias,
                                                float* __restrict__ out,
                                                int total) {
  int i = blockIdx.x * 256 + threadIdx.x;
  if (i < total) out[i] = bias[i & (DD - 1)];
}

// ---------------------------------------------------------------------------
// COO SpMM scatter: one wave per edge; lane covers 4 features.
// out[r][:] += v * x[c][:]
// ---------------------------------------------------------------------------
__global__ __launch_bounds__(256) void spmm_atomic(const int* __restrict__ rows,
                                                   const int* __restrict__ cols,
                                                   const float* __restrict__ vals,
                                                   const float* __restrict__ X,
                                                   float* __restrict__ out,
                                                   int nE) {
  const int gw = (blockIdx.x * 256 + threadIdx.x) >> 5;  // global wave id
  const int lane = threadIdx.x & 31;
  if (gw >= nE) return;

  const int r = rows[gw];
  const int c = cols[gw];
  const float v = vals[gw];

  const float4 xv =
      reinterpret_cast<const float4*>(X + (size_t)c * DD)[lane];
  float* dst = out + (size_t)r * DD + lane * 4;
  atomicAdd(dst + 0, v * xv.x);
  atomicAdd(dst + 1, v * xv.y);
  atomicAdd(dst + 2, v * xv.z);
  atomicAdd(dst + 3, v * xv.w);
}

extern "C" void kernel_launch(void* const* d_in, const int* in_sizes, int n_in,
                              void* d_out, int out_size, void* d_ws,
                              size_t ws_size, hipStream_t stream) {
  const float* X    = (const float*)d_in[0];
  const float* W1   = (const float*)d_in[1];
  const float* W2   = (const float*)d_in[2];
  const float* bias = (const float*)d_in[3];
  const float* a0v  = (const float*)d_in[4];
  const float* a1v  = (const float*)d_in[5];
  const int*   a0r  = (const int*)d_in[6];
  const int*   a0c  = (const int*)d_in[7];
  const int*   a1r  = (const int*)d_in[8];
  const int*   a1c  = (const int*)d_in[9];
  float* out = (float*)d_out;

  const int nE0 = in_sizes[4];
  const int nE1 = in_sizes[5];

  float* S1 = (float*)d_ws;                 // [N, 128]
  float* S2 = S1 + (size_t)NN * DD;         // [N, 128]

  // 1) supports via fp32 WMMA
  gemm_dual<<<NN / 16, 512, 0, stream>>>(X, W1, W2, S1, S2);

  // 2) out = bias
  const int total = NN * DD;
  init_out<<<(total + 255) / 256, 256, 0, stream>>>(bias, out, total);

  // 3) out += A0 @ S1 ; out += A1 @ S2
  spmm_atomic<<<(nE0 * 32 + 255) / 256, 256, 0, stream>>>(a0r, a0c, a0v, S1,
                                                          out, nE0);
  spmm_atomic<<<(nE1 * 32 + 255) / 256, 256, 0, stream>>>(a1r, a1c, a1v, S2,
                                                          out, nE1);
}